// SecondSelfAttention_38173669326906
// MI455X (gfx1250) — compile-verified
//
#include <hip/hip_runtime.h>
#include <hip/hip_bf16.h>
#include <math.h>

#define B_  16
#define T_  4096
#define C_  2048
#define H_  16
#define HS_ 128
#define SCALE_ 0.08838834764831845f   // 1/sqrt(128)

typedef float v2f __attribute__((ext_vector_type(2)));
typedef float v8f __attribute__((ext_vector_type(8)));

// ---------------------------------------------------------------------------
// Kernel 1/3: O[16,2048] = A[16,2048] @ W[2048,2048]^T  via V_WMMA_F32_16X16X4_F32
// 8 waves split K (256 each -> 64 WMMAs/wave), LDS reduce, 128 blocks tile N.
// ---------------------------------------------------------------------------
__global__ __launch_bounds__(256) void gemm16_wmma(const float* __restrict__ A,
                                                   const float* __restrict__ W,
                                                   float* __restrict__ O)
{
    __shared__ float red[8][8][32];
    const int lane = threadIdx.x & 31;
    const int wave = threadIdx.x >> 5;
    const int nb   = blockIdx.x * 16;          // N tile base
    const int m    = lane & 15;                // row index for A frag / B frag
    const int kg   = (lane >> 4) * 2;          // K sub-offset per half-wave
    const int k0   = wave * (C_ / 8);          // 256-wide K chunk per wave

    v8f c = {0.f, 0.f, 0.f, 0.f, 0.f, 0.f, 0.f, 0.f};

#if defined(__gfx1250__) && __has_builtin(__builtin_amdgcn_wmma_f32_16x16x4_f32)
    {
        const float* arow = A + (size_t)m * C_;
        const float* brow = W + (size_t)(nb + m) * C_;
        #pragma unroll 4
        for (int kk = k0; kk < k0 + C_ / 8; kk += 4) {
            v2f a, b;
            a.x = arow[kk + kg];
            a.y = arow[kk + kg + 1];
            b.x = brow[kk + kg];
            b.y = brow[kk + kg + 1];
            // D = A(16x4,f32) * B(4x16,f32) + C(16x16,f32)
            c = __builtin_amdgcn_wmma_f32_16x16x4_f32(
                    false, a, false, b, (short)0, c, false, false);
        }
    }
#else
    // Scalar fallback producing the identical per-lane fragment layout.
    for (int g = 0; g < 8; ++g) {
        const int mg = g + ((lane >> 4) << 3);
        const int n  = lane & 15;
        float acc = 0.f;
        for (int kk = k0; kk < k0 + C_ / 8; ++kk)
            acc += A[(size_t)mg * C_ + kk] * W[(size_t)(nb + n) * C_ + kk];
        c[g] = acc;
    }
#endif

    #pragma unroll
    for (int g = 0; g < 8; ++g) red[wave][g][lane] = c[g];
    __syncthreads();

    if (wave == 0) {
        #pragma unroll
        for (int g = 0; g < 8; ++g) {
            float s = 0.f;
            #pragma unroll
            for (int w = 0; w < 8; ++w) s += red[w][g][lane];
            const int mg = g + ((lane >> 4) << 3);   // C/D layout: M = g (+8 for hi half)
            const int n  = lane & 15;
            O[(size_t)mg * C_ + nb + n] = s;
        }
    }
}

// ---------------------------------------------------------------------------
// Kernel 2: single-token attention over the KV cache. One block per (b,h),
// 16 wave32s stream k/v with float4 (b128) coalesced loads (full 512B row
// per wave per step). Pure HBM-bandwidth phase: ~1.07 GB total.
// ---------------------------------------------------------------------------
__global__ __launch_bounds__(512) void attn_decode(const float* __restrict__ Q,
                                                   const float* __restrict__ K,
                                                   const float* __restrict__ V,
                                                   float* __restrict__ Y)
{
    __shared__ float  sc[T_];          // scores -> probabilities (16 KB)
    __shared__ float  rmax[16];
    __shared__ float  rsum[16];
    __shared__ float4 yacc[16][32];    // per-wave partial y (8 KB)

    const int lane = threadIdx.x & 31;
    const int wave = threadIdx.x >> 5;
    const int bh   = blockIdx.x;                 // b*16 + h
    const size_t base = (size_t)bh * T_ * HS_;
    const float* kp = K + base;
    const float* vp = V + base;
    const int b = bh >> 4, h = bh & 15;

    const float4 qv = *(const float4*)(Q + (size_t)b * C_ + h * HS_ + lane * 4);

    const int t0 = wave * (T_ / 16);             // 256 rows per wave
    float wmax = -INFINITY;

    // ---- phase A: scores + running max (lane owns 4 dims, butterfly reduce)
    for (int t = t0; t < t0 + T_ / 16; t += 4) {
        float4 k0v = *(const float4*)(kp + (size_t)(t + 0) * HS_ + lane * 4);
        float4 k1v = *(const float4*)(kp + (size_t)(t + 1) * HS_ + lane * 4);
        float4 k2v = *(const float4*)(kp + (size_t)(t + 2) * HS_ + lane * 4);
        float4 k3v = *(const float4*)(kp + (size_t)(t + 3) * HS_ + lane * 4);
        float d0 = qv.x * k0v.x + qv.y * k0v.y + qv.z * k0v.z + qv.w * k0v.w;
        float d1 = qv.x * k1v.x + qv.y * k1v.y + qv.z * k1v.z + qv.w * k1v.w;
        float d2 = qv.x * k2v.x + qv.y * k2v.y + qv.z * k2v.z + qv.w * k2v.w;
        float d3 = qv.x * k3v.x + qv.y * k3v.y + qv.z * k3v.z + qv.w * k3v.w;
        #pragma unroll
        for (int off = 16; off; off >>= 1) {
            d0 += __shfl_xor(d0, off, 32);
            d1 += __shfl_xor(d1, off, 32);
            d2 += __shfl_xor(d2, off, 32);
            d3 += __shfl_xor(d3, off, 32);
        }
        d0 *= SCALE_; d1 *= SCALE_; d2 *= SCALE_; d3 *= SCALE_;
        if (lane == 0) { sc[t] = d0; sc[t+1] = d1; sc[t+2] = d2; sc[t+3] = d3; }
        wmax = fmaxf(wmax, fmaxf(fmaxf(d0, d1), fmaxf(d2, d3)));
    }
    if (lane == 0) rmax[wave] = wmax;
    __syncthreads();

    float gm = -INFINITY;
    #pragma unroll
    for (int w = 0; w < 16; ++w) gm = fmaxf(gm, rmax[w]);

    // ---- phase B: exp + sum (thread-parallel over this wave's t range)
    float lsum = 0.f;
    #pragma unroll
    for (int i = 0; i < (T_ / 16) / 32; ++i) {
        const int t = t0 + i * 32 + lane;
        const float p = __expf(sc[t] - gm);
        sc[t] = p;
        lsum += p;
    }
    #pragma unroll
    for (int off = 16; off; off >>= 1) lsum += __shfl_xor(lsum, off, 32);
    if (lane == 0) rsum[wave] = lsum;
    __syncthreads();

    float tot = 0.f;
    #pragma unroll
    for (int w = 0; w < 16; ++w) tot += rsum[w];
    const float inv = 1.0f / tot;

    // ---- phase C: y = sum_t p[t] * v[t,:]  (lane owns 4 dims)
    float4 acc = {0.f, 0.f, 0.f, 0.f};
    #pragma unroll 4
    for (int t = t0; t < t0 + T_ / 16; ++t) {
        const float  p  = sc[t];                                   // LDS broadcast
        const float4 vv = *(const float4*)(vp + (size_t)t * HS_ + lane * 4);
        acc.x += p * vv.x; acc.y += p * vv.y;
        acc.z += p * vv.z; acc.w += p * vv.w;
    }
    yacc[wave][lane] = acc;
    __syncthreads();

    if (wave == 0) {
        float4 s = {0.f, 0.f, 0.f, 0.f};
        #pragma unroll
        for (int w = 0; w < 16; ++w) {
            const float4 a = yacc[w][lane];
            s.x += a.x; s.y += a.y; s.z += a.z; s.w += a.w;
        }
        s.x *= inv; s.y *= inv; s.z *= inv; s.w *= inv;
        *(float4*)(Y + (size_t)bh * HS_ + lane * 4) = s;   // Y laid out [16, 2048]
    }
}

// ---------------------------------------------------------------------------
extern "C" void kernel_launch(void* const* d_in, const int* in_sizes, int n_in,
                              void* d_out, int out_size, void* d_ws, size_t ws_size,
                              hipStream_t stream) {
    (void)in_sizes; (void)n_in; (void)out_size; (void)ws_size;
    const float* x  = (const float*)d_in[0];   // [16, 1, 2048]
    const float* k  = (const float*)d_in[1];   // [16, 16, 4096, 128]
    const float* v  = (const float*)d_in[2];   // [16, 16, 4096, 128]
    const float* Wq = (const float*)d_in[3];   // [2048, 2048]
    const float* Wp = (const float*)d_in[4];   // [2048, 2048]
    float* out = (float*)d_out;                // [16, 1, 2048]

    float* q = (float*)d_ws;                   // [16, 2048]
    float* y = q + B_ * C_;                    // [16, 2048]

    gemm16_wmma<<<C_ / 16, 256, 0, stream>>>(x, Wq, q);            // q = x @ Wq^T
    attn_decode<<<B_ * H_, 512, 0, stream>>>(q, k, v, y);          // softmax(qk)v
    gemm16_wmma<<<C_ / 16, 256, 0, stream>>>(y, Wp, out);          // out = y @ Wp^T
}